// RegisterTranslation2D_67740224192654
// MI455X (gfx1250) — compile-verified
//
#include <hip/hip_runtime.h>
#include <hip/hip_bf16.h>

typedef __attribute__((ext_vector_type(2))) float v2f;
typedef __attribute__((ext_vector_type(8))) float v8f;

#define RT_B 32
#define RT_N 512
#define RT_UF 20.0f
#define RT_REGION 30
#define RT_DFTSHIFT 15.0f

__device__ __forceinline__ float rt_ffreq(int i) {
    // fftfreq(512)*512 : [0..255, -256..-1]
    return (float)(i < 256 ? i : i - 512);
}

// ---------------------------------------------------------------------------
// 512-point Stockham radix-2 FFT in LDS. 256 threads, one butterfly each per
// stage, 9 stages, ping-pong between a and b. dir = -1 forward, +1 inverse.
// Returns pointer holding the natural-order result (fully barriered).
// ---------------------------------------------------------------------------
__device__ float2* rt_fft512(float2* a, float2* b, int t, float dir) {
    float2* src = a;
    float2* dst = b;
    for (int p = 1; p < RT_N; p <<= 1) {
        int k = t & (p - 1);
        float2 u0 = src[t];
        float2 u1 = src[t + 256];
        float ang = dir * 3.14159265358979f * (float)k / (float)p;
        float s, c;
        __sincosf(ang, &s, &c);
        float2 w = make_float2(u1.x * c - u1.y * s, u1.x * s + u1.y * c);
        int j = (t << 1) - k;
        dst[j]     = make_float2(u0.x + w.x, u0.y + w.y);
        dst[j + p] = make_float2(u0.x - w.x, u0.y - w.y);
        __syncthreads();
        float2* tmp = src; src = dst; dst = tmp;
    }
    return src;
}

// ---------------------------------------------------------------------------
// K0: init per-batch argmax slots
// ---------------------------------------------------------------------------
__global__ void rt_init_slots(unsigned long long* slots) {
    if (threadIdx.x < RT_B) slots[threadIdx.x] = 0ull;
}

// ---------------------------------------------------------------------------
// K1: forward row FFTs (real input -> complex). blockIdx.x = b*512+h,
// blockIdx.y selects target(0)/src(1). Coalesced loads/stores.
// ---------------------------------------------------------------------------
__global__ void rt_fft_rows_fwd(const float* __restrict__ tgt,
                                const float* __restrict__ src,
                                float2* __restrict__ Tf,
                                float2* __restrict__ Sf) {
    __shared__ float2 bufA[RT_N];
    __shared__ float2 bufB[RT_N];
    int t = threadIdx.x;
    int row = blockIdx.x;
    const float* in = (blockIdx.y == 0) ? tgt : src;
    float2* out = (blockIdx.y == 0) ? Tf : Sf;
    const float* r = in + (size_t)row * RT_N;
    bufA[t]       = make_float2(r[t], 0.f);
    bufA[t + 256] = make_float2(r[t + 256], 0.f);
    __syncthreads();
    float2* res = rt_fft512(bufA, bufB, t, -1.f);
    float2* o = out + (size_t)row * RT_N;
    o[t] = res[t];
    o[t + 256] = res[t + 256];
}

// ---------------------------------------------------------------------------
// K2: forward column FFTs of target & src spectra, fused cross-power
// P = S * conj(T), written IN PLACE over the target-spectrum column
// (each block owns its (b,w) column). Columns are 4KB-strided but the whole
// 134MB working set is L2-resident (192MB L2).
// ---------------------------------------------------------------------------
__global__ void rt_fft_cols_xpow(float2* __restrict__ Tf,
                                 const float2* __restrict__ Sf) {
    __shared__ float2 bufA[RT_N];
    __shared__ float2 bufB[RT_N];
    __shared__ float2 tcol[RT_N];
    int t = threadIdx.x;
    int b = blockIdx.x >> 9;
    int w = blockIdx.x & 511;
    float2* Tb = Tf + (size_t)b * RT_N * RT_N;
    const float2* Sb = Sf + (size_t)b * RT_N * RT_N;

    bufA[t]       = Tb[(size_t)t * RT_N + w];
    bufA[t + 256] = Tb[(size_t)(t + 256) * RT_N + w];
    __syncthreads();
    float2* res = rt_fft512(bufA, bufB, t, -1.f);
    tcol[t] = res[t];
    tcol[t + 256] = res[t + 256];

    bufA[t]       = Sb[(size_t)t * RT_N + w];
    bufA[t + 256] = Sb[(size_t)(t + 256) * RT_N + w];
    __syncthreads();
    res = rt_fft512(bufA, bufB, t, -1.f);

    #pragma unroll
    for (int e = 0; e < 2; ++e) {
        int h = t + e * 256;
        float2 S = res[h];
        float2 T = tcol[h];
        // S * conj(T)
        float2 P = make_float2(S.x * T.x + S.y * T.y, S.y * T.x - S.x * T.y);
        Tb[(size_t)h * RT_N + w] = P;
    }
}

// ---------------------------------------------------------------------------
// K3: inverse row FFTs of P -> Q (scale omitted: irrelevant for argmax)
// ---------------------------------------------------------------------------
__global__ void rt_ifft_rows(const float2* __restrict__ P,
                             float2* __restrict__ Q) {
    __shared__ float2 bufA[RT_N];
    __shared__ float2 bufB[RT_N];
    int t = threadIdx.x;
    int row = blockIdx.x;
    const float2* pr = P + (size_t)row * RT_N;
    bufA[t] = pr[t];
    bufA[t + 256] = pr[t + 256];
    __syncthreads();
    float2* res = rt_fft512(bufA, bufB, t, +1.f);
    float2* q = Q + (size_t)row * RT_N;
    q[t] = res[t];
    q[t + 256] = res[t + 256];
}

// ---------------------------------------------------------------------------
// K4: inverse column FFTs -> |cc|^2, per-batch argmax via packed u64
// atomicMax ((f32bits<<32)|index : strict total order -> deterministic).
// ---------------------------------------------------------------------------
__global__ void rt_ifft_cols_argmax(const float2* __restrict__ Q,
                                    unsigned long long* __restrict__ slots) {
    __shared__ float2 bufA[RT_N];
    __shared__ float2 bufB[RT_N];
    __shared__ unsigned long long red[256];
    int t = threadIdx.x;
    int b = blockIdx.x >> 9;
    int w = blockIdx.x & 511;
    const float2* Qb = Q + (size_t)b * RT_N * RT_N;
    bufA[t]       = Qb[(size_t)t * RT_N + w];
    bufA[t + 256] = Qb[(size_t)(t + 256) * RT_N + w];
    __syncthreads();
    float2* res = rt_fft512(bufA, bufB, t, +1.f);
    float2 e0 = res[t];
    float2 e1 = res[t + 256];
    float m0 = e0.x * e0.x + e0.y * e0.y;
    float m1 = e1.x * e1.x + e1.y * e1.y;
    float mb = m0;
    int hb = t;
    if (m1 > mb) { mb = m1; hb = t + 256; }
    unsigned long long pk =
        ((unsigned long long)__float_as_uint(mb) << 32) |
        (unsigned)(hb * RT_N + w);
    red[t] = pk;
    __syncthreads();
    for (int s = 128; s > 0; s >>= 1) {
        if (t < s) red[t] = (red[t] > red[t + s]) ? red[t] : red[t + s];
        __syncthreads();
    }
    if (t == 0) atomicMax(&slots[b], red[0]);
}

// ---------------------------------------------------------------------------
// K5: decode argmax -> wrapped shifts, round to 1/uf, compute DFT offsets.
// sh4 = {shift_row, shift_col, off_row, off_col}
// ---------------------------------------------------------------------------
__global__ void rt_coarse_shifts(const unsigned long long* __restrict__ slots,
                                 float4* __restrict__ sh4) {
    int b = threadIdx.x;
    if (b < RT_B) {
        unsigned idx = (unsigned)(slots[b] & 0xFFFFFFFFu);
        int row = (int)(idx >> 9);
        int col = (int)(idx & 511);
        float s0 = (row > 256) ? (float)(row - RT_N) : (float)row;
        float s1 = (col > 256) ? (float)(col - RT_N) : (float)col;
        s0 = rintf(s0 * RT_UF) / RT_UF;   // matches jnp.round (half-even)
        s1 = rintf(s1 * RT_UF) / RT_UF;
        sh4[b] = make_float4(s0, s1,
                             RT_DFTSHIFT - s0 * RT_UF,
                             RT_DFTSHIFT - s1 * RT_UF);
    }
}

// ---------------------------------------------------------------------------
// WMMA helper: D = A(16x4) * B(4x16) + C, f32
// ---------------------------------------------------------------------------
__device__ __forceinline__ v8f rt_wmma(v2f a, v2f b, v8f c) {
    return __builtin_amdgcn_wmma_f32_16x16x4_f32(
        false, a, false, b, (short)0, c, false, false);
}

// ---------------------------------------------------------------------------
// K6: stage A of upsampled DFT: Tt = conj(P) @ col_k  (512 x 32 complex).
// One wave per 16x16 output tile; grid (mt=32, nt=2, b=32). K=512 consumed
// 4 at a time with 4 v_wmma_f32_16x16x4_f32 per step (complex via 4 real
// MACs; conj folded into operand signs). col_k synthesized via __sincosf.
// A layout: lanes 0-15 hold M=lm (K=k0,k0+1), lanes 16-31 same M (K=+2).
// ---------------------------------------------------------------------------
__global__ void rt_stageA_wmma(const float2* __restrict__ P,
                               const float4* __restrict__ sh4,
                               float2* __restrict__ Tt) {
    int b = blockIdx.z, mt = blockIdx.x, nt = blockIdx.y;
    int lane = threadIdx.x;
    int half = lane >> 4;
    int lm = lane & 15;
    float off1 = sh4[b].w;
    const float2* Prow = P + (size_t)b * RT_N * RT_N + (size_t)(mt * 16 + lm) * RT_N;
    float ccol = (float)(nt * 16 + lm);
    const float ksc = 6.28318530718f / ((float)RT_N * RT_UF);
    float cm = ccol - off1;
    v8f Cr = {};
    v8f Ci = {};
    for (int kb = 0; kb < 128; ++kb) {
        int k0 = kb * 4 + half * 2;
        float2 p0 = Prow[k0];
        float2 p1 = Prow[k0 + 1];
        v2f ar  = {p0.x, p1.x};       // Re(P)
        v2f py  = {p0.y, p1.y};       // Im(P)   (= -Im(conjP))
        v2f npy = {-p0.y, -p1.y};     // -Im(P)  (=  Im(conjP))
        float s0, c0, s1, c1;
        __sincosf(ksc * rt_ffreq(k0) * cm, &s0, &c0);
        __sincosf(ksc * rt_ffreq(k0 + 1) * cm, &s1, &c1);
        v2f br = {c0, c1};            // Re(col_k) = cos
        v2f bi = {-s0, -s1};          // Im(col_k) = -sin  (exp(-i*ang))
        // conj(P)*colk: Cr += Px*br + Py*bi ; Ci += Px*bi - Py*br
        Cr = rt_wmma(ar, br, Cr);
        Cr = rt_wmma(py, bi, Cr);
        Ci = rt_wmma(ar, bi, Ci);
        Ci = rt_wmma(npy, br, Ci);
    }
    // C/D layout: N = lm, M = v + 8*half
    float2* Tb = Tt + (size_t)b * RT_N * 32;
    int c = nt * 16 + lm;
    #pragma unroll
    for (int v = 0; v < 8; ++v) {
        int hro = mt * 16 + v + 8 * half;
        Tb[(size_t)hro * 32 + c] = make_float2(Cr[v], Ci[v]);
    }
}

// ---------------------------------------------------------------------------
// K7: stage B: U = row_k @ Tt (32x32 padded, valid 30x30), |U|^2 argmax,
// final subpixel shifts -> d_out. One block (4 waves) per batch.
// ---------------------------------------------------------------------------
__global__ void rt_stageB_wmma(const float2* __restrict__ Tt,
                               const float4* __restrict__ sh4,
                               float* __restrict__ out) {
    __shared__ float svals[128];
    __shared__ int sidx[128];
    int b = blockIdx.x;
    int tid = threadIdx.x;
    int lane = tid & 31;
    int wave = tid >> 5;
    int mt = wave >> 1, nt = wave & 1;
    int half = lane >> 4;
    int lm = lane & 15;
    float4 sv = sh4[b];
    float off0 = sv.z;
    const float2* Tb = Tt + (size_t)b * RT_N * 32;
    float rrow = (float)(mt * 16 + lm) - off0;
    int c = nt * 16 + lm;
    const float ksc = 6.28318530718f / ((float)RT_N * RT_UF);
    v8f Cr = {};
    v8f Ci = {};
    for (int kb = 0; kb < 128; ++kb) {
        int k0 = kb * 4 + half * 2;
        float s0, c0, s1, c1;
        __sincosf(ksc * rrow * rt_ffreq(k0), &s0, &c0);
        __sincosf(ksc * rrow * rt_ffreq(k0 + 1), &s1, &c1);
        v2f arv  = {c0, c1};          // Re(row_k)
        v2f siv  = {s0, s1};          // -Im(row_k)
        v2f nsiv = {-s0, -s1};        //  Im(row_k)
        float2 t0 = Tb[(size_t)k0 * 32 + c];
        float2 t1 = Tb[(size_t)(k0 + 1) * 32 + c];
        v2f brv = {t0.x, t1.x};
        v2f biv = {t0.y, t1.y};
        // (cos - i sin)*(tr + i ti): Cr += cos*tr + sin*ti ; Ci += cos*ti - sin*tr
        Cr = rt_wmma(arv, brv, Cr);
        Cr = rt_wmma(siv, biv, Cr);
        Ci = rt_wmma(arv, biv, Ci);
        Ci = rt_wmma(nsiv, brv, Ci);
    }
    float bestv = -1.f;
    int besti = 0;
    #pragma unroll
    for (int v = 0; v < 8; ++v) {
        int rr = mt * 16 + v + 8 * half;
        int cc = nt * 16 + lm;
        float m = Cr[v] * Cr[v] + Ci[v] * Ci[v];
        if (rr < RT_REGION && cc < RT_REGION && m > bestv) {
            bestv = m;
            besti = rr * 32 + cc;
        }
    }
    svals[tid] = bestv;
    sidx[tid] = besti;
    __syncthreads();
    if (tid == 0) {
        float bv = -1.f;
        int bi = 0;
        for (int i = 0; i < 128; ++i)
            if (svals[i] > bv) { bv = svals[i]; bi = sidx[i]; }
        float r2 = (float)(bi >> 5) - RT_DFTSHIFT;
        float c2 = (float)(bi & 31) - RT_DFTSHIFT;
        out[b * 2 + 0] = sv.x + r2 / RT_UF;
        out[b * 2 + 1] = sv.y + c2 / RT_UF;
    }
}

// ---------------------------------------------------------------------------
// Workspace layout (bytes):
//   [0,        67108864)  R0 : target spectrum, then P (in-place)
//   [67108864, 134217728) R1 : src spectrum, then Q (rows-ifft of P)
//   [134217728,138412032) Tt : stage-A result, 32 x (512x32) complex
//   [138412032,+256)      argmax slots (u64 x 32)
//   [138412288,+512)      coarse shifts (float4 x 32)
// ---------------------------------------------------------------------------
extern "C" void kernel_launch(void* const* d_in, const int* in_sizes, int n_in,
                              void* d_out, int out_size, void* d_ws, size_t ws_size,
                              hipStream_t stream) {
    const float* tgt = (const float*)d_in[0];
    const float* src = (const float*)d_in[1];
    float* out = (float*)d_out;
    char* ws = (char*)d_ws;
    float2* R0 = (float2*)(ws);
    float2* R1 = (float2*)(ws + 67108864);
    float2* Tt = (float2*)(ws + 134217728);
    unsigned long long* slots = (unsigned long long*)(ws + 138412032);
    float4* sh4 = (float4*)(ws + 138412288);

    rt_init_slots<<<1, 32, 0, stream>>>(slots);
    rt_fft_rows_fwd<<<dim3(RT_B * RT_N, 2), 256, 0, stream>>>(tgt, src, R0, R1);
    rt_fft_cols_xpow<<<RT_B * RT_N, 256, 0, stream>>>(R0, R1);
    rt_ifft_rows<<<RT_B * RT_N, 256, 0, stream>>>(R0, R1);
    rt_ifft_cols_argmax<<<RT_B * RT_N, 256, 0, stream>>>(R1, slots);
    rt_coarse_shifts<<<1, 32, 0, stream>>>(slots, sh4);
    rt_stageA_wmma<<<dim3(32, 2, RT_B), 32, 0, stream>>>(R0, sh4, Tt);
    rt_stageB_wmma<<<RT_B, 128, 0, stream>>>(Tt, sh4, out);
}